// InPlaceABNWrapper_5669356835394
// MI455X (gfx1250) — compile-verified
//
#include <hip/hip_runtime.h>

// Shapes fixed by the reference: x[N=16, C=64, H=256, W=256] fp32, NCHW layout.
#define NB   16
#define CCH  64
#define HW   65536       // H*W
#define HW4  16384       // HW / 4 (float4 units per (n,c) segment)
#define NHW  1048576.0f  // N*H*W
#define EPS_ 1e-5f
#define SLOPE_ 0.01f

typedef __attribute__((ext_vector_type(4))) float f4;
typedef __attribute__((ext_vector_type(2))) float v2f;
typedef __attribute__((ext_vector_type(8))) float v8f;

// Full 32-lane sum using V_WMMA_F32_16X16X4_F32.
// A (16x4 f32, 2 VGPRs, documented layout): lane m (m<16) holds A[m,0]=vgpr0,
// A[m,1]=vgpr1; lane m+16 holds A[m,2]=vgpr0, A[m,3]=vgpr1.
// We set vgpr0 = per-lane partial v, vgpr1 = 0, and B = all-ones (every lane /
// VGPR = 1.0, so the B layout is irrelevant).  Then C[m,n] = v_m + v_{m+16}.
// C layout (documented): VGPR r, lanes 0-15 -> M=r, lanes 16-31 -> M=r+8.
// Summing the 8 C VGPRs per lane gives the half-sum over M; one xor-16 shuffle
// completes the full 32-lane reduction (result replicated in all lanes).
// EXEC is all-1s at every call site (no divergence in the stats kernel).
__device__ __forceinline__ float wave_sum_wmma(float v) {
  v2f a; a[0] = v;    a[1] = 0.0f;
  v2f b; b[0] = 1.0f; b[1] = 1.0f;
  v8f c = {};
  c = __builtin_amdgcn_wmma_f32_16x16x4_f32(false, a, false, b,
                                            (short)0, c, false, false);
  float r = ((c[0] + c[1]) + (c[2] + c[3])) + ((c[4] + c[5]) + (c[6] + c[7]));
  r += __shfl_xor(r, 16, 32);
  return r;
}

// ws layout (floats): [0..63]=sum, [64..127]=sumsq, [128..191]=scale, [192..255]=shift
__global__ void abn_init(float* __restrict__ acc) {
  if (threadIdx.x < 2 * CCH) acc[threadIdx.x] = 0.0f;
}

// grid = (64, 64): blockIdx.y = channel, blockIdx.x covers HW4 in 256-thread tiles.
// Each thread loads one float4 from each of the 16 batch images -> 64 floats/thread.
__global__ void abn_stats(const float* __restrict__ x, float* __restrict__ acc) {
  const int c    = blockIdx.y;
  const int seg4 = blockIdx.x * blockDim.x + threadIdx.x;  // 0..16383, exact cover
  const f4* __restrict__ xv = (const f4*)x;

  float s = 0.0f, q = 0.0f;
#pragma unroll
  for (int n = 0; n < NB; ++n) {
    f4 v = xv[(size_t)(n * CCH + c) * HW4 + seg4];
    s += (v.x + v.y) + (v.z + v.w);
    q += (v.x * v.x + v.y * v.y) + (v.z * v.z + v.w * v.w);
  }

  // wave32 reduce via WMMA, then 8-wave LDS reduce, then one atomic per block.
  float wsum = wave_sum_wmma(s);
  float wsq  = wave_sum_wmma(q);

  __shared__ float ls[8];
  __shared__ float lq[8];
  const int wave = threadIdx.x >> 5;
  if ((threadIdx.x & 31) == 0) { ls[wave] = wsum; lq[wave] = wsq; }
  __syncthreads();
  if (threadIdx.x == 0) {
    float ts = 0.0f, tq = 0.0f;
#pragma unroll
    for (int w = 0; w < 8; ++w) { ts += ls[w]; tq += lq[w]; }
    atomicAdd(&acc[c], ts);
    atomicAdd(&acc[CCH + c], tq);
  }
}

__global__ void abn_finalize(const float* __restrict__ weight,
                             const float* __restrict__ bias,
                             float* __restrict__ ws) {
  const int c = threadIdx.x;  // 64 threads
  const float mean = ws[c] * (1.0f / NHW);
  const float var  = ws[CCH + c] * (1.0f / NHW) - mean * mean;
  const float t    = var + EPS_;
  float inv = rsqrtf(t);
  inv = inv * (1.5f - 0.5f * t * inv * inv);  // one Newton step for accuracy
  const float scale = weight[c] * inv;
  ws[2 * CCH + c] = scale;
  ws[3 * CCH + c] = bias[c] - mean * scale;
}

// grid = 65536 blocks x 256 threads, one float4 per thread.
// A block spans 1024 consecutive floats, entirely inside one (n,c) segment
// (segment = 65536 floats), so the channel index is block-uniform -> scalar
// loads of scale/shift.  Output stores are non-temporal (streamed, never
// re-read) so they don't evict x lines from L2 between the two passes.
__global__ void abn_apply(const float* __restrict__ x,
                          const float* __restrict__ ws,
                          float* __restrict__ out) {
  const int  bid   = blockIdx.x;
  const int  c     = (bid >> 6) & (CCH - 1);  // 64 blocks per (n,c) segment
  const float scale = ws[2 * CCH + c];
  const float shift = ws[3 * CCH + c];

  const size_t i4 = (size_t)bid * blockDim.x + threadIdx.x;
  f4 v = __builtin_nontemporal_load(((const f4*)x) + i4);  // last use of x
  f4 y;
  y.x = fmaf(v.x, scale, shift);
  y.y = fmaf(v.y, scale, shift);
  y.z = fmaf(v.z, scale, shift);
  y.w = fmaf(v.w, scale, shift);
  y.x = (y.x >= 0.0f) ? y.x : SLOPE_ * y.x;
  y.y = (y.y >= 0.0f) ? y.y : SLOPE_ * y.y;
  y.z = (y.z >= 0.0f) ? y.z : SLOPE_ * y.z;
  y.w = (y.w >= 0.0f) ? y.w : SLOPE_ * y.w;
  __builtin_nontemporal_store(y, ((f4*)out) + i4);
}

extern "C" void kernel_launch(void* const* d_in, const int* in_sizes, int n_in,
                              void* d_out, int out_size, void* d_ws, size_t ws_size,
                              hipStream_t stream) {
  const float* x      = (const float*)d_in[0];
  const float* weight = (const float*)d_in[1];
  const float* bias   = (const float*)d_in[2];
  float*       out    = (float*)d_out;
  float*       ws     = (float*)d_ws;
  (void)in_sizes; (void)n_in; (void)out_size; (void)ws_size;

  abn_init<<<1, 128, 0, stream>>>(ws);
  abn_stats<<<dim3(64, CCH), 256, 0, stream>>>(x, ws);
  abn_finalize<<<1, CCH, 0, stream>>>(weight, bias, ws);
  abn_apply<<<65536, 256, 0, stream>>>(x, ws, out);
}